// RegimeSwitchingKF_3178275799662
// MI455X (gfx1250) — compile-verified
//
#include <hip/hip_runtime.h>
#include <hip/hip_bf16.h>

typedef float v2f __attribute__((ext_vector_type(2)));
typedef float v8f __attribute__((ext_vector_type(8)));

#define NT 64
#define OBS 9

// ---------------------------------------------------------------------------
// Stage 1: one lane per series; full regime-switching KF recursion in
// registers using Woodbury / Sherman-Morrison reductions of the structured
// innovation covariances. Writes one partial log-likelihood per block.
// ---------------------------------------------------------------------------
__global__ __launch_bounds__(256) void rskf_filter_kernel(
    const float* __restrict__ y,
    const float* __restrict__ pBG, const float* __restrict__ pBT,
    const float* __restrict__ pBB, const float* __restrict__ pBW,
    const float* __restrict__ pL1, const float* __restrict__ pL2,
    const float* __restrict__ pQ,  const float* __restrict__ pR,
    const float* __restrict__ pG1, const float* __restrict__ pG2,
    float* __restrict__ partials, int N)
{
  const float EPSV      = 1e-9f;
  const float C9        = 16.540893597684108f;   // 9*log(2*pi)
  const float LOG_1MP22 = -9.210330372476154f;   // log(1-0.9999+1e-9)
  const float LOG_P22   = -1.0000400033e-4f;     // log(0.9999+1e-9)

  const int tid = threadIdx.x;
  const int n   = blockIdx.x * blockDim.x + tid;

  // ---------------- uniform model constants ----------------
  float b0 = pBG[0], b1 = pBT[0], b2 = pBB[0], bW = pBW[0];
  float a[9];
  a[0]=1.f; a[1]=pL1[0]; a[2]=pL1[1];
  a[3]=1.f; a[4]=pL1[2]; a[5]=pL1[3];
  a[6]=1.f; a[7]=pL1[4]; a[8]=pL1[5];
  float lam[9]; lam[0]=1.f;
  #pragma unroll
  for (int o=1;o<9;++o) lam[o]=pL2[o-1];
  float q0=fabsf(pQ[0])+1e-4f, q1=fabsf(pQ[1])+1e-4f;
  float q2=fabsf(pQ[2])+1e-4f, q3=fabsf(pQ[3])+1e-4f;
  float qv[3] = {q0,q1,q2};
  float dinv[9], Rd[9];
  float logdetD = 0.f;
  #pragma unroll
  for (int o=0;o<9;++o){
    Rd[o]   = fabsf(pR[o]) + 1e-4f;     // R used in Joseph K R K^T
    float rr = Rd[o] + 1e-5f;           // R + jitter inside F
    dinv[o] = 1.f/rr;
    logdetD += __logf(rr);
  }
  float d3[3], eK[3];
  #pragma unroll
  for (int i=0;i<3;++i){
    d3[i]=0.f; eK[i]=0.f;
    #pragma unroll
    for (int k=0;k<3;++k){
      int o=3*i+k;
      d3[i] += a[o]*a[o]*dinv[o];
      eK[i] += a[o]*a[o]*dinv[o]*dinv[o]*Rd[o];
    }
  }
  // transition-12 time-invariant pieces (B21 == 0 -> Pp12 = Q always)
  float s12[3], pu12d[3];
  float logdet12 = logdetD;
  #pragma unroll
  for (int i=0;i<3;++i){
    float t = 1.f + qv[i]*d3[i];
    s12[i]  = 1.f/t;
    logdet12 += __logf(t);
    pu12d[i] = s12[i]*s12[i]*qv[i]*(1.f + qv[i]*eK[i]);
  }
  // rank-1 (regime-2 observation) constants
  float u2[9], sig2=0.f, h0=0.f;
  #pragma unroll
  for (int o=0;o<9;++o){
    u2[o] = lam[o]*dinv[o];
    sig2 += lam[o]*u2[o];
    h0   += u2[o]*u2[o]*Rd[o];
  }
  float g1=pG1[0], g20=pG2[0], g21c=pG2[1], g22c=pG2[2];

  // ---------------- per-series state ----------------
  float et[3] = {0.f,0.f,0.f};                      // regime-1 mean (comp 3 always 0)
  float P00=1000.f,P01=0.f,P02=0.f,P11=1000.f,P12=0.f,P22=1000.f; // regime-1 cov 3x3
  float m2=0.f, p2v=1e-9f;                          // regime-2 mean/cov (comp 3 only)
  float pr1=0.99f, pr2=0.01f;
  float llacc = 0.f;

  if (n < N) {
    const float* yp = y + (size_t)n * (NT*OBS);
    for (int t=0;t<NT;++t){
      float yv[9];
      #pragma unroll
      for (int o=0;o<9;++o) yv[o]=yp[t*9+o];

      // ================= transition 1->1 =================
      float e11[3] = {b0*et[0], b1*et[1], b2*et[2]};
      float Pb[3][3];
      Pb[0][0]=b0*b0*P00+q0; Pb[0][1]=b0*b1*P01;    Pb[0][2]=b0*b2*P02;
      Pb[1][0]=Pb[0][1];     Pb[1][1]=b1*b1*P11+q1; Pb[1][2]=b1*b2*P12;
      Pb[2][0]=Pb[0][2];     Pb[2][1]=Pb[1][2];     Pb[2][2]=b2*b2*P22+q2;
      float u[3]={0.f,0.f,0.f}; float t1a=0.f;
      #pragma unroll
      for (int i=0;i<3;++i){
        #pragma unroll
        for (int k=0;k<3;++k){
          int o=3*i+k;
          float v = yv[o] - a[o]*e11[i];
          u[i] += a[o]*dinv[o]*v;
          t1a  += dinv[o]*v*v;
        }
      }
      // G = I + Pb*diag(d3)  (Woodbury capacity, 3x3)
      float G[3][3];
      #pragma unroll
      for (int i=0;i<3;++i)
        #pragma unroll
        for (int j=0;j<3;++j) G[i][j]=((i==j)?1.f:0.f)+Pb[i][j]*d3[j];
      float c00=G[1][1]*G[2][2]-G[1][2]*G[2][1];
      float c01=G[1][0]*G[2][2]-G[1][2]*G[2][0];
      float c02=G[1][0]*G[2][1]-G[1][1]*G[2][0];
      float detG=G[0][0]*c00-G[0][1]*c01+G[0][2]*c02;
      float rdet=1.f/detG;
      float Gi[3][3];
      Gi[0][0]=  c00*rdet;
      Gi[0][1]= -(G[0][1]*G[2][2]-G[0][2]*G[2][1])*rdet;
      Gi[0][2]=  (G[0][1]*G[1][2]-G[0][2]*G[1][1])*rdet;
      Gi[1][0]= -c01*rdet;
      Gi[1][1]=  (G[0][0]*G[2][2]-G[0][2]*G[2][0])*rdet;
      Gi[1][2]= -(G[0][0]*G[1][2]-G[0][2]*G[1][0])*rdet;
      Gi[2][0]=  c02*rdet;
      Gi[2][1]= -(G[0][0]*G[2][1]-G[0][1]*G[2][0])*rdet;
      Gi[2][2]=  (G[0][0]*G[1][1]-G[0][1]*G[1][0])*rdet;
      float M3[3][3];
      #pragma unroll
      for (int i=0;i<3;++i)
        #pragma unroll
        for (int j=0;j<3;++j)
          M3[i][j]=Gi[i][0]*Pb[0][j]+Gi[i][1]*Pb[1][j]+Gi[i][2]*Pb[2][j];
      float mu[3];
      #pragma unroll
      for (int i=0;i<3;++i) mu[i]=M3[i][0]*u[0]+M3[i][1]*u[1]+M3[i][2]*u[2];
      float quad11 = t1a - (u[0]*mu[0]+u[1]*mu[1]+u[2]*mu[2]);
      float ll11 = -0.5f*(quad11 + logdetD + __logf(detG) + C9);
      float eu11[3] = {e11[0]+mu[0], e11[1]+mu[1], e11[2]+mu[2]};
      // Joseph form: Pu = (I-M3*D3) Pb (I-M3*D3)^T + M3 diag(eK) M3^T
      float Jm[3][3];
      #pragma unroll
      for (int i=0;i<3;++i)
        #pragma unroll
        for (int j=0;j<3;++j) Jm[i][j]=((i==j)?1.f:0.f)-M3[i][j]*d3[j];
      float Tm[3][3];
      #pragma unroll
      for (int i=0;i<3;++i)
        #pragma unroll
        for (int j=0;j<3;++j)
          Tm[i][j]=Jm[i][0]*Pb[0][j]+Jm[i][1]*Pb[1][j]+Jm[i][2]*Pb[2][j];
      float Pu11[3][3];
      #pragma unroll
      for (int i=0;i<3;++i)
        #pragma unroll
        for (int j=i;j<3;++j)
          Pu11[i][j]=Tm[i][0]*Jm[j][0]+Tm[i][1]*Jm[j][1]+Tm[i][2]*Jm[j][2]
                    +M3[i][0]*eK[0]*M3[j][0]+M3[i][1]*eK[1]*M3[j][1]+M3[i][2]*eK[2]*M3[j][2];

      // ================= transition 2->1 (e=0, Pp=Q; F const) =================
      float u12[3]={0.f,0.f,0.f}; float t1b=0.f;
      #pragma unroll
      for (int i=0;i<3;++i){
        #pragma unroll
        for (int k=0;k<3;++k){
          int o=3*i+k;
          u12[i]+=a[o]*dinv[o]*yv[o];
          t1b  +=dinv[o]*yv[o]*yv[o];
        }
      }
      float quad12 = t1b - (qv[0]*s12[0]*u12[0]*u12[0]
                           +qv[1]*s12[1]*u12[1]*u12[1]
                           +qv[2]*s12[2]*u12[2]*u12[2]);
      float ll12 = -0.5f*(quad12 + logdet12 + C9);
      float eu12[3] = {q0*s12[0]*u12[0], q1*s12[1]*u12[1], q2*s12[2]*u12[2]};

      // ================= transition 1->2 (rank-1 F) =================
      float e21  = (et[0]+et[1]+et[2])*(1.f/3.f);
      float ssum = P00+P11+P22 + 2.f*(P01+P02+P12);
      float p33a = q3 + ssum*(1.f/9.f);
      float t1c=0.f, t2c=0.f;
      #pragma unroll
      for (int o=0;o<9;++o){ float v=yv[o]-lam[o]*e21; t1c+=dinv[o]*v*v; t2c+=u2[o]*v; }
      float dena   = 1.f + p33a*sig2;
      float quad21 = t1c - p33a*t2c*t2c/dena;
      float ll21   = -0.5f*(quad21 + logdetD + __logf(dena) + C9);
      float eu21   = e21 + p33a*t2c/dena;
      float pu21   = (p33a + p33a*p33a*h0)/(dena*dena);

      // ================= transition 2->2 (rank-1 F) =================
      float e22  = bW*m2;
      float p33b = q3 + bW*bW*p2v;
      float t1d=0.f, t2d=0.f;
      #pragma unroll
      for (int o=0;o<9;++o){ float v=yv[o]-lam[o]*e22; t1d+=dinv[o]*v*v; t2d+=u2[o]*v; }
      float denb   = 1.f + p33b*sig2;
      float quad22 = t1d - p33b*t2d*t2d/denb;
      float ll22   = -0.5f*(quad22 + logdetD + __logf(denb) + C9);
      float eu22   = e22 + p33b*t2d/denb;
      float pu22   = (p33b + p33b*p33b*h0)/(denb*denb);

      // ================= HMM mixing =================
      float logit = g1 + g20*et[0] + g21c*et[1] + g22c*et[2];
      float p11   = 1.f/(1.f+__expf(-logit));
      float lp1 = __logf(pr1+EPSV), lp2 = __logf(pr2+EPSV);
      float c11v = lp1 + __logf(p11+EPSV)      + ll11;
      float c12v = lp2 + LOG_1MP22             + ll12;
      float c21v = lp1 + __logf(1.f-p11+EPSV)  + ll21;
      float c22v = lp2 + LOG_P22               + ll22;
      float mx = fmaxf(fmaxf(c11v,c12v),fmaxf(c21v,c22v));
      float x11=__expf(c11v-mx), x12=__expf(c12v-mx);
      float x21=__expf(c21v-mx), x22=__expf(c22v-mx);
      float se   = x11+x12+x21+x22;
      float ll_t = mx + __logf(se);
      float rse  = 1.f/se;
      float pt11=x11*rse, pt12=x12*rse, pt21=x21*rse, pt22=x22*rse;
      float npr1 = pt11+pt12, npr2 = pt21+pt22;
      float r1 = 1.f/(npr1+EPSV), r2 = 1.f/(npr2+EPSV);
      float W11=pt11*r1, W12=pt12*r1, W21=pt21*r2, W22=pt22*r2;

      // collapse regime 1 (Pu12 is diagonal)
      float ne[3], da[3], db[3];
      #pragma unroll
      for (int i=0;i<3;++i){
        ne[i]=W11*eu11[i]+W12*eu12[i];
        da[i]=eu11[i]-ne[i]; db[i]=eu12[i]-ne[i];
      }
      float nP00=W11*(Pu11[0][0]+da[0]*da[0])+W12*(pu12d[0]+db[0]*db[0]);
      float nP01=W11*(Pu11[0][1]+da[0]*da[1])+W12*(db[0]*db[1]);
      float nP02=W11*(Pu11[0][2]+da[0]*da[2])+W12*(db[0]*db[2]);
      float nP11=W11*(Pu11[1][1]+da[1]*da[1])+W12*(pu12d[1]+db[1]*db[1]);
      float nP12=W11*(Pu11[1][2]+da[1]*da[2])+W12*(db[1]*db[2]);
      float nP22=W11*(Pu11[2][2]+da[2]*da[2])+W12*(pu12d[2]+db[2]*db[2]);
      // collapse regime 2 (scalar)
      float nm2=W21*eu21+W22*eu22;
      float dc=eu21-nm2, dd=eu22-nm2;
      float np2=W21*(pu21+dc*dc)+W22*(pu22+dd*dd);

      et[0]=ne[0]; et[1]=ne[1]; et[2]=ne[2];
      P00=nP00;P01=nP01;P02=nP02;P11=nP11;P12=nP12;P22=nP22;
      m2=nm2; p2v=np2; pr1=npr1; pr2=npr2;
      llacc += ll_t;
    }
  }

  // deterministic block reduction
  __shared__ float red[256];
  red[tid] = llacc;
  __syncthreads();
  #pragma unroll
  for (int s=128;s>0;s>>=1){
    if (tid < s) red[tid]+=red[tid+s];
    __syncthreads();
  }
  if (tid==0) partials[blockIdx.x]=red[0];
}

// ---------------------------------------------------------------------------
// Stage 2: single-wave WMMA reduction of block partials.
// A (16x4 f32, 2 VGPRs) carries 64 partial sums per iteration, B = ones,
// C accumulates row sums. D[m][n] = rowsum_m for every n, so each lane's 8
// accumulator regs hold half the row sums; one shfl finishes the total.
// The grand total is independent of the exact A-operand lane striping.
// ---------------------------------------------------------------------------
__global__ void rskf_reduce_kernel(const float* __restrict__ partials,
                                   float* __restrict__ out, int nb)
{
  int l = threadIdx.x;
  v8f C = {0.f,0.f,0.f,0.f,0.f,0.f,0.f,0.f};
  v2f Bv; Bv.x = 1.f; Bv.y = 1.f;
  for (int base = 0; base < nb; base += 64){
    v2f Av;
    int i0 = base + l, i1 = base + 32 + l;
    Av.x = (i0 < nb) ? partials[i0] : 0.f;
    Av.y = (i1 < nb) ? partials[i1] : 0.f;
    // v_wmma_f32_16x16x4_f32: D = A(16x4) * B(4x16) + C
    C = __builtin_amdgcn_wmma_f32_16x16x4_f32(false, Av, false, Bv,
                                              (short)0, C, false, false);
  }
  float p = C[0]+C[1]+C[2]+C[3]+C[4]+C[5]+C[6]+C[7];
  float other = __shfl_xor(p, 16, 32);   // lanes 0..15 hold rowsums 0-7, 16..31 hold 8-15
  if (l == 0) out[0] = p + other;
}

extern "C" void kernel_launch(void* const* d_in, const int* in_sizes, int n_in,
                              void* d_out, int out_size, void* d_ws, size_t ws_size,
                              hipStream_t stream)
{
  const float* y   = (const float*)d_in[0];
  const float* bg  = (const float*)d_in[1];
  const float* bt  = (const float*)d_in[2];
  const float* bb  = (const float*)d_in[3];
  const float* bw  = (const float*)d_in[4];
  const float* l1f = (const float*)d_in[5];
  const float* l2f = (const float*)d_in[6];
  const float* qd  = (const float*)d_in[7];
  const float* rd  = (const float*)d_in[8];
  const float* g1  = (const float*)d_in[9];
  const float* g2  = (const float*)d_in[10];

  int N = in_sizes[0] / (NT*OBS);
  int blocks = (N + 255) / 256;
  float* partials = (float*)d_ws;

  rskf_filter_kernel<<<blocks, 256, 0, stream>>>(
      y, bg, bt, bb, bw, l1f, l2f, qd, rd, g1, g2, partials, N);
  rskf_reduce_kernel<<<1, 32, 0, stream>>>(partials, (float*)d_out, blocks);
}